// HyperbolicAttentionBias_77936476554060
// MI455X (gfx1250) — compile-verified
//
#include <hip/hip_runtime.h>

typedef __attribute__((ext_vector_type(2))) float v2f;
typedef __attribute__((ext_vector_type(8))) float v8f;

#define NTOK  30522
#define DIM   64
#define BATCH 4
#define SEQ   1024
#define LN2   0.69314718055994531f

// -------- Kernel 1: gather + Poincare-ball projection --------
// One thread per token. Writes projected embedding, squared norm, and the
// reciprocal 1/(1 - ||x||^2) so the tile epilogue needs no divides.
__global__ void __launch_bounds__(256)
poincare_gather_project(const int* __restrict__ ids,
                        const float* __restrict__ weight,
                        float* __restrict__ emb,
                        float* __restrict__ x2,
                        float* __restrict__ rinv) {
  int t = blockIdx.x * blockDim.x + threadIdx.x;
  if (t >= BATCH * SEQ) return;
  int id = ids[t];
  id = min(max(id, 0), NTOK - 1);
  const float4* w = (const float4*)(weight + (size_t)id * DIM);

  float4 v[16];
  float ss = 0.0f;
#pragma unroll
  for (int i = 0; i < 16; ++i) {
    v[i] = w[i];
    ss += v[i].x * v[i].x + v[i].y * v[i].y + v[i].z * v[i].z + v[i].w * v[i].w;
  }
  float norm = sqrtf(ss);
  const float max_norm = 1.0f - 1e-5f;          // (1 - PROJ_EPS)/sqrt(C), C=1
  float f = (norm > max_norm) ? (max_norm / fmaxf(norm, 1e-12f)) : 1.0f;

  float4* e = (float4*)(emb + (size_t)t * DIM);
#pragma unroll
  for (int i = 0; i < 16; ++i) {
    float4 o;
    o.x = v[i].x * f; o.y = v[i].y * f; o.z = v[i].z * f; o.w = v[i].w * f;
    e[i] = o;
  }
  float nf = norm * f;
  float n2 = nf * nf;                            // < 1 by projection
  x2[t]   = n2;
  rinv[t] = 1.0f / (1.0f - n2);
}

// -------- Kernel 2: fp32 WMMA Gram tile + fast acosh epilogue --------
// One wave32 per upper-triangle 16x16 tile (ti <= tj); symmetry gives the
// mirrored tile via an LDS transpose, halving WMMA + epilogue work.
__global__ void __launch_bounds__(256)
poincare_bias_tile(const float* __restrict__ emb,
                   const float* __restrict__ x2,
                   const float* __restrict__ rinv,
                   const float* __restrict__ scale_p,
                   float* __restrict__ out) {
  __shared__ float tbuf[8][16 * 17];            // per-wave padded tile

  const int wave = threadIdx.x >> 5;
  const int lane = threadIdx.x & 31;
  const int t    = blockIdx.x * 8 + wave;       // tile id in [0, 16384)
  const int b    = t >> 12;                     // batch
  const int rem  = t & 4095;
  const int ti   = rem >> 6;                    // row-tile (M/16)
  const int tj   = rem & 63;                    // col-tile (N/16)
  if (ti > tj) return;                          // wave-uniform: EXEC stays ~0

  // Fragment layout for V_WMMA_F32_16X16X4_F32 (ISA 7.12.2):
  //   A 16x4:  lanes 0-15 -> M=lane, VGPR0=K0,VGPR1=K1; lanes 16-31 -> K2,K3
  //   B 4x16:  mirrored with lane -> N
  const int idx = lane & 15;                    // M for A-frag, N for B-frag
  const int h   = lane >> 4;                    // K-pair / M-half selector
  const float* rowA = emb + ((size_t)(b * SEQ + ti * 16 + idx)) * DIM + h * 2;
  const float* rowB = emb + ((size_t)(b * SEQ + tj * 16 + idx)) * DIM + h * 2;

  v8f c = {};
#pragma unroll
  for (int k = 0; k < 16; ++k) {
    v2f a  = *(const v2f*)(rowA + k * 4);
    v2f bb = *(const v2f*)(rowB + k * 4);
    c = __builtin_amdgcn_wmma_f32_16x16x4_f32(false, a, false, bb,
                                              (short)0, c, false, false);
  }

  // Per-lane epilogue constants. C/D layout: N = lane%16, M = v + 8*(lane/16).
  const float factor = -(*scale_p) * LN2;       // fold ln2 (v_log_f32 is log2)
  const int jcol = tj * 16 + (lane & 15);
  const float xj = x2[b * SEQ + jcol];
  const float rj = rinv[b * SEQ + jcol];
  // x2/rinv for the 8 contiguous rows i = ti*16 + 8*h + v
  const float4* x2p = (const float4*)(x2   + b * SEQ + ti * 16 + 8 * h);
  const float4* rp  = (const float4*)(rinv + b * SEQ + ti * 16 + 8 * h);
  float4 xiv[2] = {x2p[0], x2p[1]};
  float4 riv[2] = {rp[0],  rp[1]};
  const float xi_a[8] = {xiv[0].x, xiv[0].y, xiv[0].z, xiv[0].w,
                         xiv[1].x, xiv[1].y, xiv[1].z, xiv[1].w};
  const float ri_a[8] = {riv[0].x, riv[0].y, riv[0].z, riv[0].w,
                         riv[1].x, riv[1].y, riv[1].z, riv[1].w};

  float* outb = out + ((size_t)b << 20);        // b * SEQ * SEQ
  float* lds  = tbuf[wave];
#pragma unroll
  for (int v = 0; v < 8; ++v) {
    const int m = v + 8 * h;                    // tile-local row
    const int i = ti * 16 + m;
    const float xi = xi_a[v];
    // acosh(1+t) = ln(1 + t + sqrt(t*(t+2))), t = 2*||x-y||^2 / denom
    float sq = fmaxf(xi + xj - 2.0f * c[v], 0.0f);
    float tt = fmaxf(2.0f * sq * (ri_a[v] * rj), 1e-7f);   // arg clamp
    float s  = tt + __builtin_amdgcn_sqrtf(tt * (tt + 2.0f));
    float d  = factor * __builtin_amdgcn_logf(1.0f + s);   // log2 * ln2
    outb[(size_t)i * SEQ + jcol] = d;
    lds[m * 17 + (lane & 15)] = d;              // stage for transpose
  }

  if (ti != tj) {
    // Mirrored tile (tj,ti): element (r,cc) = original (cc,r) from LDS.
    // Same-wave DS ops are in-order; stride-17 reads are bank-conflict-free.
#pragma unroll
    for (int v = 0; v < 8; ++v) {
      const int r  = v + 8 * h;                 // mirrored tile-local row
      const int cc = lane & 15;                 // mirrored tile-local col
      float d = lds[cc * 17 + r];
      outb[(size_t)(tj * 16 + r) * SEQ + ti * 16 + cc] = d;
    }
  }
}

extern "C" void kernel_launch(void* const* d_in, const int* in_sizes, int n_in,
                              void* d_out, int out_size, void* d_ws, size_t ws_size,
                              hipStream_t stream) {
  const int*   ids    = (const int*)d_in[0];
  const float* weight = (const float*)d_in[1];
  const float* scale  = (const float*)d_in[2];

  float* emb  = (float*)d_ws;                       // 4*1024*64 floats = 1 MB
  float* x2   = emb + (size_t)BATCH * SEQ * DIM;    // 4096 floats
  float* rinv = x2 + (size_t)BATCH * SEQ;           // 4096 floats

  poincare_gather_project<<<(BATCH * SEQ + 255) / 256, 256, 0, stream>>>(
      ids, weight, emb, x2, rinv);

  // 4 batches * 64*64 tiles = 16384 wave-slots, 8 waves/block -> 2048 blocks.
  // Lower-triangle waves (ti > tj) exit immediately; ti <= tj waves write
  // their tile and its transpose.
  poincare_bias_tile<<<2048, 256, 0, stream>>>(emb, x2, rinv, scale,
                                               (float*)d_out);
}